// SE2ClassEmbeddings_7310034337920
// MI455X (gfx1250) — compile-verified
//
#include <hip/hip_runtime.h>
#include <hip/hip_bf16.h>

// ---------------------------------------------------------------------------
// SE2 class embeddings:
//   cls[b,n,e,c]   = sum_t w[e,t] * cls_in[b,n,t,c]              (complex)
//   out[b,l,c,x,y] = sum_{m,e} x[b,m,e,x,y] * cls[b,(l-m)%8,e,c] (complex)
// B=8, MAXM=8, E=64, T=128, NCLS=10, X=Y=64.
// Both einsums implemented as fp32 WMMA (V_WMMA_F32_16X16X4_F32) GEMMs.
// ---------------------------------------------------------------------------

typedef __attribute__((ext_vector_type(2))) float v2f;
typedef __attribute__((ext_vector_type(8))) float v8f;

#if defined(__HIP_DEVICE_COMPILE__)
#if !__has_builtin(__builtin_amdgcn_wmma_f32_16x16x4_f32)
#error "wmma_f32_16x16x4_f32 builtin not available on device"
#endif
#endif

static __device__ __forceinline__ v8f wmma4(v2f a, v2f b, v8f c) {
    // 8 args: (neg_a, A, neg_b, B, c_mod, C, reuse_a, reuse_b)
    return __builtin_amdgcn_wmma_f32_16x16x4_f32(false, a, false, b,
                                                 (short)0, c, false, false);
}

#define NB    8      // batch
#define NM    8      // MAXM
#define NE    64     // E
#define NT    128    // T
#define NC    10     // NCLS
#define NXY   4096   // X*Y
#define NK    512    // M*E contraction length
#define LDA   516    // padded LDS row stride (floats), keeps 8B align, kills bank conflicts

// ---------------------------------------------------------------------------
// Kernel 1: cls[b,n,e,c] = sum_t w[e,t] * cls_in[b,n,t,c]   (complex, planar out)
// One block per (b,n); 4 waves, each owns one 16-row e-tile. M=64,K=128,N=10(pad16).
// B columns 10..15 read a clamped (duplicated) column: they only affect D
// columns 10..15, which are never stored -> branchless, EXEC stays all-1s.
// ---------------------------------------------------------------------------
__global__ __launch_bounds__(128) void se2_cls_proj_kernel(
    const float* __restrict__ wr, const float* __restrict__ wi,
    const float* __restrict__ cr, const float* __restrict__ ci,
    float* __restrict__ out_re,   float* __restrict__ out_im)
{
    const int bn   = blockIdx.x;            // b*8 + n
    const int wave = threadIdx.x >> 5;      // e-tile 0..3
    const int lane = threadIdx.x & 31;
    const int row  = lane & 15;
    const int hi   = lane >> 4;             // 0: K pair (k,k+1), 1: (k+2,k+3)
    const int e    = wave * 16 + row;
    const int col  = lane & 15;             // c (valid when < 10)
    const int ccol = (col < NC) ? col : (NC - 1);   // clamp, no predication

    const float* __restrict__ crb = cr + (size_t)bn * NT * NC;
    const float* __restrict__ cib = ci + (size_t)bn * NT * NC;

    v8f acc_re = {};
    v8f acc_im = {};

#pragma unroll 4
    for (int t0 = 0; t0 < NT; t0 += 4) {
        const int ka = t0 + hi * 2;
        // A fragment: rows of W, two consecutive t -> contiguous 8B load
        v2f a_re = *(const v2f*)&wr[e * NT + ka];
        v2f a_im = *(const v2f*)&wi[e * NT + ka];
        // B fragment: cls_in[t, c] (clamped col, unconditional loads)
        v2f b_re, b_im;
        b_re.x = crb[(ka    ) * NC + ccol];
        b_re.y = crb[(ka + 1) * NC + ccol];
        b_im.x = cib[(ka    ) * NC + ccol];
        b_im.y = cib[(ka + 1) * NC + ccol];
        v2f a_nim = -a_im;
        acc_re = wmma4(a_re,  b_re, acc_re);
        acc_re = wmma4(a_nim, b_im, acc_re);
        acc_im = wmma4(a_re,  b_im, acc_im);
        acc_im = wmma4(a_im,  b_re, acc_im);
    }

    // D layout: VGPR j -> row j + 8*hi, col = lane&15
    if (col < NC) {
#pragma unroll
        for (int j = 0; j < 8; ++j) {
            const int eo = wave * 16 + j + hi * 8;
            const size_t o = ((size_t)bn * NE + eo) * NC + col;
            out_re[o] = acc_re[j];
            out_im[o] = acc_im[j];
        }
    }
}

// ---------------------------------------------------------------------------
// Kernel 2: out[b,l,c,xy] = sum_k A[(l,c),k] * x[b,k,xy],  k = m*64+e,
//           A[(l,c),(m,e)] = cls[b,(l-m)%8,e,c]  (complex)
// Grid: (16, 5, 8). Block = 256 (8 waves). Each wave owns TWO adjacent
// 16x16 N-tiles (32 xy columns), reusing each A fragment for 8 WMMAs/k-step
// (1024 WMMAs per wave over K=512). Block stages its gathered 16x512 complex
// A-slice in LDS once, shared by all 8 waves.
// ---------------------------------------------------------------------------
__global__ __launch_bounds__(256) void se2_main_gemm_kernel(
    const float* __restrict__ xr, const float* __restrict__ xi,
    const float* __restrict__ clsr, const float* __restrict__ clsi,
    float* __restrict__ out)
{
    __shared__ float Ar[16 * LDA];
    __shared__ float Ai[16 * LDA];

    const int b   = blockIdx.z;
    const int mt  = blockIdx.y;             // 0..4, rows 16*mt .. 16*mt+15 of 80
    const int tid = threadIdx.x;

    // ---- stage gathered A slice (circulant index (l-m)%8) into LDS ----
    for (int i = tid; i < 16 * NK; i += 256) {
        const int r = i >> 9;               // local row 0..15
        const int k = i & (NK - 1);         // 0..511
        const int R = mt * 16 + r;          // global row 0..79
        const int l = R / NC, c = R % NC;
        const int m = k >> 6, e = k & 63;
        const int n = (l - m) & 7;
        const size_t src = ((size_t)(b * NM + n) * NE + e) * NC + c;
        Ar[r * LDA + k] = clsr[src];
        Ai[r * LDA + k] = clsi[src];
    }
    __syncthreads();

    const int wave  = tid >> 5;
    const int lane  = tid & 31;
    const int row   = lane & 15;            // A row / D col within tile
    const int hi    = lane >> 4;
    const int n0    = (blockIdx.x * 8 + wave) * 32;  // base xy column (2 tiles)
    const int col   = lane & 15;

    const float* __restrict__ xr_b = xr + (size_t)b * NK * NXY;
    const float* __restrict__ xi_b = xi + (size_t)b * NK * NXY;

    v8f acc_re0 = {}, acc_im0 = {};
    v8f acc_re1 = {}, acc_im1 = {};

#pragma unroll 2
    for (int k0 = 0; k0 < NK; k0 += 4) {
        const int ka = k0 + hi * 2;
        // A fragment from LDS (8B-aligned, padded stride -> conflict-free)
        v2f a_re = *(const v2f*)&Ar[row * LDA + ka];
        v2f a_im = *(const v2f*)&Ai[row * LDA + ka];
        v2f a_nim = -a_im;
        // B fragments from global x: 32 consecutive floats per k-plane
        const size_t p0 = (size_t)(ka    ) * NXY + n0 + col;
        const size_t p1 = (size_t)(ka + 1) * NXY + n0 + col;
        v2f br0, bi0, br1, bi1;
        br0.x = xr_b[p0];       br0.y = xr_b[p1];
        bi0.x = xi_b[p0];       bi0.y = xi_b[p1];
        br1.x = xr_b[p0 + 16];  br1.y = xr_b[p1 + 16];
        bi1.x = xi_b[p0 + 16];  bi1.y = xi_b[p1 + 16];

        acc_re0 = wmma4(a_re,  br0, acc_re0);
        acc_re0 = wmma4(a_nim, bi0, acc_re0);   // re -= A_im*B_im
        acc_im0 = wmma4(a_re,  bi0, acc_im0);
        acc_im0 = wmma4(a_im,  br0, acc_im0);

        acc_re1 = wmma4(a_re,  br1, acc_re1);
        acc_re1 = wmma4(a_nim, bi1, acc_re1);
        acc_im1 = wmma4(a_re,  bi1, acc_im1);
        acc_im1 = wmma4(a_im,  br1, acc_im1);
    }

    // ---- store: complex64 interleaved (re,im) -> contiguous float2 per lane ----
#pragma unroll
    for (int j = 0; j < 8; ++j) {
        const int R = mt * 16 + j + hi * 8;     // 80 rows = exactly 5 tiles, no mask
        const int l = R / NC, c = R % NC;
        const size_t o = ((((size_t)b * NM + l) * NC + c) * NXY + n0 + col) * 2;
        *(float2*)&out[o]      = make_float2(acc_re0[j], acc_im0[j]);
        *(float2*)&out[o + 32] = make_float2(acc_re1[j], acc_im1[j]);
    }
}

extern "C" void kernel_launch(void* const* d_in, const int* in_sizes, int n_in,
                              void* d_out, int out_size, void* d_ws, size_t ws_size,
                              hipStream_t stream) {
    const float* x_real   = (const float*)d_in[0];
    const float* x_imag   = (const float*)d_in[1];
    const float* cls_real = (const float*)d_in[2];
    const float* cls_imag = (const float*)d_in[3];
    const float* w_real   = (const float*)d_in[4];
    const float* w_imag   = (const float*)d_in[5];
    float* out = (float*)d_out;

    // workspace: planar projected cls, re + im (B*MAXM*E*NCLS floats each)
    float* ws_re = (float*)d_ws;                    // 40960 floats
    float* ws_im = ws_re + NB * NM * NE * NC;       // 40960 floats

    se2_cls_proj_kernel<<<dim3(NB * NM), dim3(128), 0, stream>>>(
        w_real, w_imag, cls_real, cls_imag, ws_re, ws_im);

    se2_main_gemm_kernel<<<dim3(NXY / (16 * 2 * 8), 5, NB), dim3(256), 0, stream>>>(
        x_real, x_imag, ws_re, ws_im, out);
}